// Custom_45011257262495
// MI455X (gfx1250) — compile-verified
//
#include <hip/hip_runtime.h>

typedef float v2f __attribute__((ext_vector_type(2)));
typedef float v8f __attribute__((ext_vector_type(8)));

#define HID   64
#define NEDGE 1600000
#define NUSER 100000
#define NITEM 100000

// LDS row-pair stride (floats). 160 mod 64 == 32, so the two half-waves of a
// ds_load_b64 B-fragment fetch land on disjoint bank halves -> conflict-free.
#define WPAIR_STRIDE 160

// ---------------------------------------------------------------- zero fill
__global__ void zero_kernel(float* __restrict__ p, int n) {
    int i = blockIdx.x * blockDim.x + threadIdx.x;
    if (i < n) p[i] = 0.0f;
}

// ---------------------------------------------------------------- edge scatter
// One wave (32 lanes) per edge: coalesced 128B+128B row read of h_src[src],
// per-lane f32 atomic adds into agg[dst], lane0 bumps cnt[dst].
// Working set (h, agg) fits in MI455X's 192MB L2 -> atomics stay on-chip.
__global__ __launch_bounds__(256) void scatter_kernel(
    const float* __restrict__ h, const int* __restrict__ edge,
    float* __restrict__ agg, float* __restrict__ cnt, int nEdge)
{
    int gw   = blockIdx.x * 8 + (threadIdx.x >> 5);   // global wave id = edge id
    int lane = threadIdx.x & 31;
    if (gw >= nEdge) return;
    int src = edge[gw];
    int dst = edge[nEdge + gw];
    const float* hs = h + (size_t)src * HID;
    float* ad       = agg + (size_t)dst * HID;
    atomicAdd(ad + lane,       hs[lane]);
    atomicAdd(ad + lane + 32,  hs[lane + 32]);
    if (lane == 0) atomicAdd(cnt + dst, 1.0f);
}

// ---------------------------------------------------------------- fused GEMM
// out[M x 64] = relu( (A / max(cnt,1)) @ W1[K1 x 64] + bias + H @ W2[64 x 64] )
// cnt == nullptr -> no mean division; H/W2 == nullptr -> skip second GEMM.
// 256 threads = 8 waves; each wave computes a 16x64 tile via 4 accumulators
// of v_wmma_f32_16x16x4_f32.
//
// f32 WMMA fragment layout (ISA 7.12.2):
//   A 16x4 : lane = {m = lane&15}; VGPR0/1 = K = 2*(lane>>4) + {0,1}
//   B 4x16 : lane = {n = lane&15}; VGPR0/1 = K = 2*(lane>>4) + {0,1}
//   C 16x16: VGPR r -> row r + 8*(lane>>4), col = lane&15
//
// W is staged in LDS pair-swizzled: element (k, n) lives at
//   lds[(k>>1)*WPAIR_STRIDE + 2*n + (k&1)]
// so a lane's B fragment {W[k][n], W[k+1][n]} is one aligned ds_load_b64,
// eliminating the register-shuffle movs the row-major layout forced.
__global__ __launch_bounds__(256) void gemm_kernel(
    const float* __restrict__ A,  const float* __restrict__ cnt,
    const float* __restrict__ W1, const float* __restrict__ bias,
    const float* __restrict__ H,  const float* __restrict__ W2,
    float* __restrict__ out, int M, int K1)
{
    // W1: up to 64 k-pairs, W2: 32 k-pairs, stride 160 -> 15360 floats = 60KB
    __shared__ float lds[(64 + 32) * WPAIR_STRIDE];
    float* ldsW1 = lds;
    float* ldsW2 = lds + 64 * WPAIR_STRIDE;

    int tid = threadIdx.x;
    for (int i = tid; i < K1 * HID; i += 256) {
        int k = i >> 6, n = i & 63;
        ldsW1[(k >> 1) * WPAIR_STRIDE + 2 * n + (k & 1)] = W1[i];
    }
    if (W2) {
        for (int i = tid; i < HID * HID; i += 256) {
            int k = i >> 6, n = i & 63;
            ldsW2[(k >> 1) * WPAIR_STRIDE + 2 * n + (k & 1)] = W2[i];
        }
    }
    __syncthreads();

    int wave = tid >> 5;
    int lane = tid & 31;
    int l15  = lane & 15;
    int hi   = lane >> 4;                 // which K-pair of the 4-wide K slab
    int khi  = hi << 1;                   // 0 or 2
    int m0   = blockIdx.x * 128 + wave * 16;
    if (m0 >= M) return;                  // uniform per wave; EXEC stays all-ones

    v8f acc0 = {}, acc1 = {}, acc2 = {}, acc3 = {};

    float scale = 1.0f;
    if (cnt) scale = 1.0f / fmaxf(cnt[m0 + l15], 1.0f);

    const float* Arow = A + (size_t)(m0 + l15) * K1;
    for (int k = 0; k < K1; k += 4) {
        v2f a;
        a.x = Arow[k + khi]     * scale;
        a.y = Arow[k + khi + 1] * scale;
        const float* Bk = ldsW1 + ((k >> 1) + hi) * WPAIR_STRIDE + 2 * l15;
        v2f b0 = *(const v2f*)(Bk +  0);
        v2f b1 = *(const v2f*)(Bk + 32);
        v2f b2 = *(const v2f*)(Bk + 64);
        v2f b3 = *(const v2f*)(Bk + 96);
        acc0 = __builtin_amdgcn_wmma_f32_16x16x4_f32(false, a, false, b0, (short)0, acc0, false, false);
        acc1 = __builtin_amdgcn_wmma_f32_16x16x4_f32(false, a, false, b1, (short)0, acc1, false, false);
        acc2 = __builtin_amdgcn_wmma_f32_16x16x4_f32(false, a, false, b2, (short)0, acc2, false, false);
        acc3 = __builtin_amdgcn_wmma_f32_16x16x4_f32(false, a, false, b3, (short)0, acc3, false, false);
    }

    if (H) {
        const float* Hrow = H + (size_t)(m0 + l15) * HID;
        for (int k = 0; k < HID; k += 4) {
            v2f a;
            a.x = Hrow[k + khi];
            a.y = Hrow[k + khi + 1];
            const float* Bk = ldsW2 + ((k >> 1) + hi) * WPAIR_STRIDE + 2 * l15;
            v2f b0 = *(const v2f*)(Bk +  0);
            v2f b1 = *(const v2f*)(Bk + 32);
            v2f b2 = *(const v2f*)(Bk + 64);
            v2f b3 = *(const v2f*)(Bk + 96);
            acc0 = __builtin_amdgcn_wmma_f32_16x16x4_f32(false, a, false, b0, (short)0, acc0, false, false);
            acc1 = __builtin_amdgcn_wmma_f32_16x16x4_f32(false, a, false, b1, (short)0, acc1, false, false);
            acc2 = __builtin_amdgcn_wmma_f32_16x16x4_f32(false, a, false, b2, (short)0, acc2, false, false);
            acc3 = __builtin_amdgcn_wmma_f32_16x16x4_f32(false, a, false, b3, (short)0, acc3, false, false);
        }
    }

    // epilogue: bias + relu, C layout: VGPR r -> row m0 + r + 8*(lane>>4)
    int rowhi = hi << 3;
    float b0v = bias[l15], b1v = bias[16 + l15], b2v = bias[32 + l15], b3v = bias[48 + l15];
    for (int r = 0; r < 8; ++r) {
        int row = m0 + r + rowhi;
        float* o = out + (size_t)row * HID + l15;
        o[0]  = fmaxf(acc0[r] + b0v, 0.0f);
        o[16] = fmaxf(acc1[r] + b1v, 0.0f);
        o[32] = fmaxf(acc2[r] + b2v, 0.0f);
        o[48] = fmaxf(acc3[r] + b3v, 0.0f);
    }
}

// ---------------------------------------------------------------- launcher
extern "C" void kernel_launch(void* const* d_in, const int* in_sizes, int n_in,
                              void* d_out, int out_size, void* d_ws, size_t ws_size,
                              hipStream_t stream) {
    (void)in_sizes; (void)n_in; (void)out_size; (void)ws_size;
    const float* x_user  = (const float*)d_in[0];
    const float* x_item  = (const float*)d_in[1];
    const int*   edge_ui = (const int*)d_in[2];
    const int*   edge_iu = (const int*)d_in[3];
    const float* Wu      = (const float*)d_in[4];
    const float* bu      = (const float*)d_in[5];
    const float* Wi      = (const float*)d_in[6];
    const float* bi      = (const float*)d_in[7];
    const float* Wl      = (const float*)d_in[8];
    const float* bl      = (const float*)d_in[9];
    const float* Wr      = (const float*)d_in[10];

    float* ws  = (float*)d_ws;
    float* hu  = ws;                               // [NUSER x 64]
    float* hi  = hu + (size_t)NUSER * HID;         // [NITEM x 64]
    float* nu  = hi + (size_t)NITEM * HID;         // layer-0 out (user)
    float* ni  = nu + (size_t)NUSER * HID;         // layer-0 out (item)
    float* agg = ni + (size_t)NITEM * HID;         // [max(N) x 64]
    float* cnt = agg + (size_t)NUSER * HID;        // [max(N)]
    float* outp = (float*)d_out;

    dim3 blk(256);
    const int gemmGridU = (NUSER + 127) / 128;
    const int gemmGridI = (NITEM + 127) / 128;
    const int edgeGrid  = (NEDGE + 7) / 8;         // 8 waves/block, 1 wave/edge

    // input linears: h = relu(x @ W + b)
    gemm_kernel<<<gemmGridU, blk, 0, stream>>>(x_user, nullptr, Wu, bu, nullptr, nullptr, hu, NUSER, 128);
    gemm_kernel<<<gemmGridI, blk, 0, stream>>>(x_item, nullptr, Wi, bi, nullptr, nullptr, hi, NITEM, 64);

    for (int l = 0; l < 2; ++l) {
        const float* hu_in = (l == 0) ? hu : nu;
        const float* hi_in = (l == 0) ? hi : ni;
        float* out_u = (l == 0) ? nu : outp;                          // hu first in d_out
        float* out_i = (l == 0) ? ni : (outp + (size_t)NUSER * HID);  // then hi
        const float* Wl0 = Wl + ((size_t)l * 2 + 0) * HID * HID;
        const float* Wl1 = Wl + ((size_t)l * 2 + 1) * HID * HID;
        const float* Wr0 = Wr + ((size_t)l * 2 + 0) * HID * HID;
        const float* Wr1 = Wr + ((size_t)l * 2 + 1) * HID * HID;
        const float* bl0 = bl + ((size_t)l * 2 + 0) * HID;
        const float* bl1 = bl + ((size_t)l * 2 + 1) * HID;

        // edge type 0: user -> item
        zero_kernel<<<(NITEM * HID + 255) / 256, blk, 0, stream>>>(agg, NITEM * HID);
        zero_kernel<<<(NITEM + 255) / 256,       blk, 0, stream>>>(cnt, NITEM);
        scatter_kernel<<<edgeGrid, blk, 0, stream>>>(hu_in, edge_ui, agg, cnt, NEDGE);
        gemm_kernel<<<gemmGridI, blk, 0, stream>>>(agg, cnt, Wl0, bl0, hi_in, Wr0, out_i, NITEM, 64);

        // edge type 1: item -> user
        zero_kernel<<<(NUSER * HID + 255) / 256, blk, 0, stream>>>(agg, NUSER * HID);
        zero_kernel<<<(NUSER + 255) / 256,       blk, 0, stream>>>(cnt, NUSER);
        scatter_kernel<<<edgeGrid, blk, 0, stream>>>(hi_in, edge_iu, agg, cnt, NEDGE);
        gemm_kernel<<<gemmGridU, blk, 0, stream>>>(agg, cnt, Wl1, bl1, hu_in, Wr1, out_u, NUSER, 64);
    }
}